// MyConv2D_7292854468609
// MI455X (gfx1250) — compile-verified
//
#include <hip/hip_runtime.h>
#include <hip/hip_bf16.h>

// C(F=64, N=116250) = W(64x128) x X^T(128x116250), all f32, via
// V_WMMA_F32_16X16X4_F32. Memory-bound (~90 MB @ 23.3 TB/s ~= 3.8 us);
// f32 WMMA keeps full reference precision.

typedef __attribute__((ext_vector_type(2))) float v2f;
typedef __attribute__((ext_vector_type(8))) float v8f;

#define N_TOTAL 116250
#define DDIM 128
#define NT_PER_WAVE 2                      // 16-wide n-tiles per wave
#define WAVES_PER_BLOCK 8
#define N_PER_BLOCK (WAVES_PER_BLOCK * NT_PER_WAVE * 16)   // 256 n per block

__global__ __launch_bounds__(256) void conv_gemm_wmma_f32(
    const float* __restrict__ x,   // (N_TOTAL, 128)
    const float* __restrict__ w,   // (64, 128)
    float* __restrict__ out)       // (64, N_TOTAL) row-major
{
    const int lane  = threadIdx.x & 31;
    const int wave  = threadIdx.x >> 5;
    const int lrow  = lane & 15;            // row (A) / col (B) within tile
    const int khalf = (lane >> 4) << 1;     // lanes 0-15 -> K+0, lanes 16-31 -> K+2

    const long long n_base =
        (long long)blockIdx.x * N_PER_BLOCK + (long long)wave * (NT_PER_WAVE * 16);

    v8f acc[4][NT_PER_WAVE];
    {
        v8f zero = {};
        #pragma unroll
        for (int ft = 0; ft < 4; ++ft)
            #pragma unroll
            for (int nt = 0; nt < NT_PER_WAVE; ++nt)
                acc[ft][nt] = zero;
    }

    // Per-lane base pointers. x rows clamped so EXEC stays all-1s for WMMA.
    const float* xrow[NT_PER_WAVE];
    #pragma unroll
    for (int nt = 0; nt < NT_PER_WAVE; ++nt) {
        long long n = n_base + nt * 16 + lrow;
        if (n > (long long)(N_TOTAL - 1)) n = N_TOTAL - 1;
        xrow[nt] = x + n * DDIM + khalf;
    }
    const float* wrow = w + (long long)lrow * DDIM + khalf;  // + ft*16*DDIM per f-tile

    for (int kb = 0; kb < DDIM; kb += 4) {
        v2f a[4], b[NT_PER_WAVE];
        #pragma unroll
        for (int ft = 0; ft < 4; ++ft)
            a[ft] = *(const v2f*)(wrow + ft * 16 * DDIM + kb);   // A: 16x4 of W
        #pragma unroll
        for (int nt = 0; nt < NT_PER_WAVE; ++nt)
            b[nt] = *(const v2f*)(xrow[nt] + kb);                // B: 4x16 of X^T
        #pragma unroll
        for (int ft = 0; ft < 4; ++ft)
            #pragma unroll
            for (int nt = 0; nt < NT_PER_WAVE; ++nt)
                acc[ft][nt] = __builtin_amdgcn_wmma_f32_16x16x4_f32(
                    /*neg_a=*/false, a[ft],
                    /*neg_b=*/false, b[nt],
                    /*c_mod=*/(short)0, acc[ft][nt],
                    /*reuse_a=*/false, /*reuse_b=*/false);
    }

    // D layout: VGPR r -> M = r (lanes 0-15) or r+8 (lanes 16-31); N = lane%16.
    const int mhalf = (lane >> 4) << 3;     // 0 or 8
    #pragma unroll
    for (int nt = 0; nt < NT_PER_WAVE; ++nt) {
        const long long n = n_base + nt * 16 + lrow;
        if (n < (long long)N_TOTAL) {
            #pragma unroll
            for (int ft = 0; ft < 4; ++ft) {
                float* orow = out + (long long)(ft * 16 + mhalf) * N_TOTAL + n;
                #pragma unroll
                for (int r = 0; r < 8; ++r)
                    orow[(long long)r * N_TOTAL] = acc[ft][nt][r];
            }
        }
    }
}

extern "C" void kernel_launch(void* const* d_in, const int* in_sizes, int n_in,
                              void* d_out, int out_size, void* d_ws, size_t ws_size,
                              hipStream_t stream) {
    const float* x = (const float*)d_in[0];   // (1,1,N,128) f32
    const float* w = (const float*)d_in[1];   // (64,1,1,128) f32
    float* out = (float*)d_out;               // (5,64,62,375) == (64, N) row-major view

    const int grid = (N_TOTAL + N_PER_BLOCK - 1) / N_PER_BLOCK;  // 455
    conv_gemm_wmma_f32<<<grid, 256, 0, stream>>>(x, w, out);
}